// REM_83992380441451
// MI455X (gfx1250) — compile-verified
//
#include <hip/hip_runtime.h>
#include <cmath>

// Toeplitz-band generator for MI455X (gfx1250, wave32).
//
// out[i][j][h] depends only on d = trunc(|i-j|) in [0,200] and 10 scalars ->
// 201x8 fp32 LDS table (6.4 KB) per block. Store-bandwidth bound: 134 MB out
// (fits the 192 MB L2; RT stores ack at L2).
//
// Row layout: [constant | 401-pt band | constant]; constants are table[d=0].
// Data plane: CDNA5 async LDS->global stores (GLOBAL_STORE_ASYNC_FROM_LDS_B128,
// ASYNCcnt) — waves only compute addresses; data goes LDS -> memory directly
// without transiting VGPRs. Fallback: ds_load_b128 + global_store_b128.

typedef float v4f __attribute__((ext_vector_type(4)));
typedef int   v4i __attribute__((vector_size(16)));   // matches builtin pointee

#define NHEAD 8
#define TRUNC_D 200
#define TSIZE ((TRUNC_D + 1) * NHEAD)   // 1608 floats = 6432 bytes LDS
#define ROWS_PER_BLOCK 4
#define BLOCK_THREADS 256

#if defined(__has_builtin)
#  if __has_builtin(__builtin_amdgcn_global_store_async_from_lds_b128)
#    define USE_ASYNC_STORES 1
#  endif
#endif
#ifndef USE_ASYNC_STORES
#  define USE_ASYNC_STORES 0
#endif

typedef __attribute__((address_space(1))) v4i as1_v4i;   // global (__device__)
typedef __attribute__((address_space(3))) v4i as3_v4i;   // LDS (__shared__)

__global__ __launch_bounds__(BLOCK_THREADS)
void toeplitz_gen_kernel(const float* __restrict__ p_eta,
                         const float* __restrict__ p_nu,
                         const float* __restrict__ p_theta,
                         float* __restrict__ out,
                         int Q, int K)
{
    __shared__ float table[TSIZE] __attribute__((aligned(16)));

    // ---- Phase 1: build the 201x8 value table (fp64 pow for accuracy; ~7
    //      entries/thread once per block, negligible vs the store stream).
    const float lam = tanhf(p_eta[0]);
    const float gam = 1.0f / (1.0f + __expf(-p_nu[0]));

    for (int idx = threadIdx.x; idx < TSIZE; idx += BLOCK_THREADS) {
        const int d = idx >> 3;       // 0..200
        const int h = idx & 7;
        const float df = (float)d;
        float v;
        if (h == 4) {
            v = (float)pow((double)lam, (double)df);                 // lam^d
        } else if (h < 6) {
            const float g = (float)pow((double)gam, (double)df);     // gam^d
            const float m = df * p_theta[h];
            v = (h == 2 || h == 3) ? g * sinf(m) : g * cosf(m);
        } else {
            float L;                                                 // dilated
            if (h == 6) L = ((d & 1) == 0) ? df * 0.5f : 0.0f;
            else        L = ((d % 3) == 0) ? df * (1.0f / 3.0f) : 0.0f;
            const float g = (float)pow((double)gam, (double)L);
            v = g * sinf(L * p_theta[h]);
        }
        table[idx] = v;
    }
    __syncthreads();

    const v4f* __restrict__ tab4 = (const v4f*)table;
#if !USE_ASYNC_STORES
    const v4f c0 = tab4[0];   // d=0 pattern, heads 0..3
    const v4f c1 = tab4[1];   // d=0 pattern, heads 4..7
#endif

    // ---- Phase 2: stream rows. One row = 2*K float4 slots; thread t owns
    //      slots t, t+256, ... so each wave32 store covers 512 contiguous B.
    const int row0 = blockIdx.x * ROWS_PER_BLOCK;
    const int nvec = K * 2;

    // Emit float4 slots [fs, fe) of the constant d=0 pattern.
    auto store_const = [&](v4f* __restrict__ dst, int fs, int fe) {
        for (int f = fs + (int)threadIdx.x; f < fe; f += BLOCK_THREADS) {
#if USE_ASYNC_STORES
            __builtin_amdgcn_global_store_async_from_lds_b128(
                (as1_v4i*)(dst + f), (as3_v4i*)(tab4 + (f & 1)), 0, 0);
#else
            dst[f] = (f & 1) ? c1 : c0;   // parity is loop-invariant per thread
#endif
        }
    };

    for (int r = 0; r < ROWS_PER_BLOCK; ++r) {
        const int i = row0 + r;
        if (i >= Q) break;
        v4f* __restrict__ dst = (v4f*)(out + (size_t)i * (size_t)K * NHEAD);

        int jlo = i - TRUNC_D; if (jlo < 0) jlo = 0;
        int jhi = i + TRUNC_D; if (jhi > K - 1) jhi = K - 1;
        const int f0 = 2 * jlo;          // band slots [f0, f1)
        const int f1 = 2 * (jhi + 1);

        store_const(dst, 0, f0);         // left out-of-band region

        for (int f = f0 + (int)threadIdx.x; f < f1; f += BLOCK_THREADS) {
            const int j = f >> 1;
            int d = i - j;
            d = (d < 0) ? -d : d;        // guaranteed <= 200 inside band
#if USE_ASYNC_STORES
            __builtin_amdgcn_global_store_async_from_lds_b128(
                (as1_v4i*)(dst + f), (as3_v4i*)(tab4 + (d * 2 + (f & 1))), 0, 0);
#else
            dst[f] = tab4[d * 2 + (f & 1)];
#endif
        }

        store_const(dst, f1, nvec);      // right out-of-band region
    }

#if USE_ASYNC_STORES
    // s_endpgm implicitly waits all counters, but make completion explicit so
    // the LDS table provably outlives all in-flight async reads.
    __builtin_amdgcn_s_wait_asynccnt(0);
#endif
}

extern "C" void kernel_launch(void* const* d_in, const int* in_sizes, int n_in,
                              void* d_out, int out_size, void* d_ws, size_t ws_size,
                              hipStream_t stream) {
    (void)in_sizes; (void)n_in; (void)d_ws; (void)ws_size;
    const float* eta   = (const float*)d_in[0];
    const float* nu    = (const float*)d_in[1];
    const float* theta = (const float*)d_in[2];
    // d_in[3]/d_in[4] hold query_len/key_len on-device; derive Q=K from
    // out_size (out = Q*K*8 with Q==K in this problem).
    const long long npts = (long long)out_size / NHEAD;
    const int Q = (int)(sqrt((double)npts) + 0.5);
    const int K = Q;

    const int nblocks = (Q + ROWS_PER_BLOCK - 1) / ROWS_PER_BLOCK;
    toeplitz_gen_kernel<<<dim3(nblocks), dim3(BLOCK_THREADS), 0, stream>>>(
        eta, nu, theta, (float*)d_out, Q, K);
}